// NodeTransformerConvLayer_74045236183750
// MI455X (gfx1250) — compile-verified
//
#include <hip/hip_runtime.h>

#define NN 50000
#define NE 800000
#define INSZ 256
#define OUTC 128
#define EDIM 64
#define EPB 64      // edges per block in edge passes
#define EPAD 132    // padded LDS row stride (floats), keeps 16B alignment

typedef float v2f __attribute__((ext_vector_type(2)));
typedef float v8f __attribute__((ext_vector_type(8)));

__device__ __forceinline__ v8f wmma_f32_k4(v2f a, v2f b, v8f c) {
    // V_WMMA_F32_16X16X4_F32 : D = A(16x4,f32) * B(4x16,f32) + C(16x16,f32)
    return __builtin_amdgcn_wmma_f32_16x16x4_f32(false, a, false, b, (short)0, c,
                                                 false, false);
}

// float atomic max via sign-split integer atomics (IEEE monotonic trick)
__device__ __forceinline__ void atomicMaxF32(float* addr, float val) {
    if (val >= 0.0f) atomicMax((int*)addr, __float_as_int(val));
    else             atomicMin((unsigned int*)addr, __float_as_uint(val));
}

// ---------------------------------------------------------------- init
__global__ void init_kernel(float* __restrict__ out, float* __restrict__ m,
                            float* __restrict__ s) {
    int i = blockIdx.x * blockDim.x + threadIdx.x;
    if (i < NN * OUTC) out[i] = 0.0f;
    if (i < NN) { m[i] = -__builtin_inff(); s[i] = 0.0f; }
}

// ------------------------------------------------- node projections (WMMA)
// grid = (NN/16, 4), block = 256 (8 waves). Wave w computes 16x16 tile at
// columns [16w, 16w+16) of matrix blockIdx.y in {Q,K,V,S}.
__global__ __launch_bounds__(256)
void proj_kernel(const float* __restrict__ x,
                 const float* __restrict__ Wq, const float* __restrict__ bq, float* Q,
                 const float* __restrict__ Wk, const float* __restrict__ bk, float* K,
                 const float* __restrict__ Wv, const float* __restrict__ bv, float* V,
                 const float* __restrict__ Ws, const float* __restrict__ bs, float* S) {
    const float *W, *b; float* O;
    switch (blockIdx.y) {
        case 0:  W = Wq; b = bq; O = Q; break;
        case 1:  W = Wk; b = bk; O = K; break;
        case 2:  W = Wv; b = bv; O = V; break;
        default: W = Ws; b = bs; O = S; break;
    }
    const int wave = threadIdx.x >> 5;
    const int lane = threadIdx.x & 31;
    const int half = lane >> 4;      // 0: K pair {0,1}, 1: K pair {2,3}
    const int l    = lane & 15;
    const int m0   = blockIdx.x * 16;
    const int n0   = wave * 16;
    const float* arow = x + (size_t)(m0 + l) * INSZ;
    v8f acc = {};
    for (int k0 = 0; k0 < INSZ; k0 += 4) {
        const int ka = k0 + 2 * half;
        v2f a;  a.x  = arow[ka];                       a.y  = arow[ka + 1];
        v2f bf; bf.x = W[(size_t)ka * OUTC + n0 + l];  bf.y = W[(size_t)(ka + 1) * OUTC + n0 + l];
        acc = wmma_f32_k4(a, bf, acc);
    }
    const float bias = b[n0 + l];
    for (int i = 0; i < 8; ++i)
        O[(size_t)(m0 + i + 8 * half) * OUTC + n0 + l] = acc[i] + bias;
}

// ------------------------- WMMA edge-projection tile: e = edge_attr@We + be
// 8 waves: wave w covers edges [(w>>1)*16, +16) x cols [(w&1)*64, +64)
__device__ __forceinline__ void edge_proj_tile(const float* __restrict__ edge_attr,
                                               const float* __restrict__ We,
                                               const float* __restrict__ be,
                                               int e0, float* __restrict__ elds) {
    const int wave = threadIdx.x >> 5;
    const int lane = threadIdx.x & 31;
    const int half = lane >> 4;
    const int l    = lane & 15;
    const int es   = (wave >> 1) * 16;   // edge strip within block
    const int ch   = (wave & 1) * 64;    // column half
    v8f acc0 = {}, acc1 = {}, acc2 = {}, acc3 = {};
    const float* arow = edge_attr + (size_t)(e0 + es + l) * EDIM;
    for (int k0 = 0; k0 < EDIM; k0 += 4) {
        const int ka = k0 + 2 * half;
        v2f a; a.x = arow[ka]; a.y = arow[ka + 1];
        const float* w0 = We + (size_t)ka * OUTC;
        const float* w1 = We + (size_t)(ka + 1) * OUTC;
        v2f b0; b0.x = w0[ch + l];      b0.y = w1[ch + l];
        v2f b1; b1.x = w0[ch + 16 + l]; b1.y = w1[ch + 16 + l];
        v2f b2; b2.x = w0[ch + 32 + l]; b2.y = w1[ch + 32 + l];
        v2f b3; b3.x = w0[ch + 48 + l]; b3.y = w1[ch + 48 + l];
        acc0 = wmma_f32_k4(a, b0, acc0);
        acc1 = wmma_f32_k4(a, b1, acc1);
        acc2 = wmma_f32_k4(a, b2, acc2);
        acc3 = wmma_f32_k4(a, b3, acc3);
    }
    for (int t = 0; t < 4; ++t) {
        v8f acc = (t == 0) ? acc0 : (t == 1) ? acc1 : (t == 2) ? acc2 : acc3;
        const int n0 = ch + 16 * t;
        const float bias = be[n0 + l];
        for (int i = 0; i < 8; ++i)
            elds[(es + i + 8 * half) * EPAD + n0 + l] = acc[i] + bias;
    }
}

// ------------------------------------------- pass 1: alpha + scatter-max
__global__ __launch_bounds__(256)
void edge_pass1(const float* __restrict__ edge_attr, const float* __restrict__ We,
                const float* __restrict__ be, const int* __restrict__ ei,
                const float* __restrict__ Q, const float* __restrict__ Km,
                float* __restrict__ alpha, float* __restrict__ mmax) {
    __shared__ __align__(16) float elds[EPB * EPAD];
    const int e0 = blockIdx.x * EPB;
    edge_proj_tile(edge_attr, We, be, e0, elds);
    __syncthreads();
    const int t  = threadIdx.x;
    const int eL = t >> 2, p = t & 3;     // 4 lanes per edge, 32 cols each
    const int ge  = e0 + eL;
    const int src = ei[ge];
    const int dst = ei[NE + ge];
    const float4* qr = (const float4*)(Q    + (size_t)dst * OUTC + p * 32);
    const float4* kr = (const float4*)(Km   + (size_t)src * OUTC + p * 32);
    const float4* er = (const float4*)(elds + eL * EPAD + p * 32);
    float part = 0.0f;
    for (int j = 0; j < 8; ++j) {
        float4 q4 = qr[j], k4 = kr[j], e4 = er[j];
        part += q4.x * (k4.x + e4.x) + q4.y * (k4.y + e4.y)
              + q4.z * (k4.z + e4.z) + q4.w * (k4.w + e4.w);
    }
    part += __shfl_xor(part, 1);
    part += __shfl_xor(part, 2);
    if (p == 0) {
        const float aval = part * 0.08838834764831845f;  // 1/sqrt(128)
        alpha[ge] = aval;
        atomicMaxF32(&mmax[dst], aval);
    }
}

// ----------------------------------- exp(alpha - m) and scatter-sum denom
__global__ void edge_sum(const int* __restrict__ ei, float* __restrict__ alpha,
                         const float* __restrict__ mmax, float* __restrict__ ssum) {
    int e = blockIdx.x * blockDim.x + threadIdx.x;
    if (e >= NE) return;
    const int dst = ei[NE + e];
    const float a = __expf(alpha[e] - mmax[dst]);
    alpha[e] = a;
    atomicAdd(&ssum[dst], a);
}

// --------------------------------- pass 2: weighted message scatter-add
__global__ __launch_bounds__(256)
void edge_pass2(const float* __restrict__ edge_attr, const float* __restrict__ We,
                const float* __restrict__ be, const int* __restrict__ ei,
                const float* __restrict__ V, const float* __restrict__ alpha,
                const float* __restrict__ ssum, float* __restrict__ out) {
    __shared__ __align__(16) float elds[EPB * EPAD];
    const int e0 = blockIdx.x * EPB;
    edge_proj_tile(edge_attr, We, be, e0, elds);
    __syncthreads();
    const int t  = threadIdx.x;
    const int eL = t >> 2, p = t & 3;
    const int ge  = e0 + eL;
    const int src = ei[ge];
    const int dst = ei[NE + ge];
    const float coeff = alpha[ge] / (ssum[dst] + 1e-16f);
    const float4* vr = (const float4*)(V    + (size_t)src * OUTC + p * 32);
    const float4* er = (const float4*)(elds + eL * EPAD + p * 32);
    float* orow = out + (size_t)dst * OUTC + p * 32;
    for (int j = 0; j < 8; ++j) {
        float4 v4 = vr[j], e4 = er[j];
        atomicAdd(orow + 4 * j + 0, (v4.x + e4.x) * coeff);
        atomicAdd(orow + 4 * j + 1, (v4.y + e4.y) * coeff);
        atomicAdd(orow + 4 * j + 2, (v4.z + e4.z) * coeff);
        atomicAdd(orow + 4 * j + 3, (v4.w + e4.w) * coeff);
    }
}

// ---------------------------------------------------- skip + ELU epilogue
__global__ void finish_kernel(float* __restrict__ out, const float* __restrict__ S) {
    int i = blockIdx.x * blockDim.x + threadIdx.x;
    if (i >= NN * OUTC) return;
    const float o = out[i] + S[i];
    out[i] = (o > 0.0f) ? o : (__expf(o) - 1.0f);
}

extern "C" void kernel_launch(void* const* d_in, const int* in_sizes, int n_in,
                              void* d_out, int out_size, void* d_ws, size_t ws_size,
                              hipStream_t stream) {
    const float* x         = (const float*)d_in[0];
    const int*   ei        = (const int*)d_in[1];      // [2, NE]
    const float* edge_attr = (const float*)d_in[2];
    const float* Wq = (const float*)d_in[3];  const float* bq = (const float*)d_in[4];
    const float* Wk = (const float*)d_in[5];  const float* bk = (const float*)d_in[6];
    const float* Wv = (const float*)d_in[7];  const float* bv = (const float*)d_in[8];
    const float* We = (const float*)d_in[9];  const float* be = (const float*)d_in[10];
    const float* Ws = (const float*)d_in[11]; const float* bs = (const float*)d_in[12];
    float* out = (float*)d_out;

    // workspace layout: Q | K | V | S | alpha | m | s   (~106 MB)
    float* Q     = (float*)d_ws;
    float* Kp    = Q  + (size_t)NN * OUTC;
    float* Vp    = Kp + (size_t)NN * OUTC;
    float* Sp    = Vp + (size_t)NN * OUTC;
    float* alpha = Sp + (size_t)NN * OUTC;
    float* mmax  = alpha + NE;
    float* ssum  = mmax + NN;

    init_kernel<<<(NN * OUTC + 255) / 256, 256, 0, stream>>>(out, mmax, ssum);
    proj_kernel<<<dim3(NN / 16, 4), 256, 0, stream>>>(x, Wq, bq, Q, Wk, bk, Kp,
                                                      Wv, bv, Vp, Ws, bs, Sp);
    edge_pass1<<<NE / EPB, 256, 0, stream>>>(edge_attr, We, be, ei, Q, Kp, alpha, mmax);
    edge_sum<<<(NE + 255) / 256, 256, 0, stream>>>(ei, alpha, mmax, ssum);
    edge_pass2<<<NE / EPB, 256, 0, stream>>>(edge_attr, We, be, ei, Vp, alpha, ssum, out);
    finish_kernel<<<(NN * OUTC + 255) / 256, 256, 0, stream>>>(out, Sp);
}